// CumulativeLayerNorm_33483565039653
// MI455X (gfx1250) — compile-verified
//
#include <hip/hip_runtime.h>
#include <hip/hip_bf16.h>

typedef __attribute__((ext_vector_type(2))) float v2f;
typedef __attribute__((ext_vector_type(8))) float v8f;

#define EPSV 1e-6f

// ---------------------------------------------------------------------------
// Kernel 1: per-(b,t) channel sums via fp32 WMMA.
// Each wave32 owns a 16-wide t-tile; loops c in steps of 4.
// A = ones(16x4), B = x[c..c+3, t0..t0+15]  =>  D row0 = sum over channels.
// Second accumulator takes B*B for the sum of squares.
// Addressing is strength-reduced: per-lane base pointer, bumped by 4*T.
// ---------------------------------------------------------------------------
__global__ void k_colsum_wmma(const float* __restrict__ x,
                              float* __restrict__ S,
                              float* __restrict__ Q,
                              int C, int T) {
    const int wave = threadIdx.x >> 5;       // 4 waves per block
    const int lane = threadIdx.x & 31;
    const int tilesPerB = T >> 4;            // T/16 tiles per batch
    const int tile = blockIdx.x * 4 + wave;  // grid sized exactly
    const int b  = tile / tilesPerB;
    const int t0 = (tile % tilesPerB) << 4;

    const int n  = lane & 15;                // N column (time)
    const int kh = lane >> 4;                // 0 -> K rows {0,1}, 1 -> {2,3}

    const size_t strideT = (size_t)T;
    // lane-constant base: row (2*kh) of the first K-block, column t0+n
    const float* p = x + (size_t)b * C * strideT
                       + (size_t)(2 * kh) * strideT
                       + (size_t)t0 + n;
    const size_t bump = 4 * strideT;         // advance 4 channels per step

    v2f ones; ones[0] = 1.0f; ones[1] = 1.0f;
    v8f accS = {};
    v8f accQ = {};

    #pragma unroll 4
    for (int c0 = 0; c0 < C; c0 += 4) {
        const float b0 = p[0];
        const float b1 = p[strideT];
        p += bump;
        v2f Bv; Bv[0] = b0;      Bv[1] = b1;
        v2f Bq; Bq[0] = b0 * b0; Bq[1] = b1 * b1;
        // 8 args: (neg_a, A, neg_b, B, c_mod, C, reuse_a, reuse_b)
        accS = __builtin_amdgcn_wmma_f32_16x16x4_f32(false, ones, false, Bv,
                                                     (short)0, accS, false, false);
        accQ = __builtin_amdgcn_wmma_f32_16x16x4_f32(false, ones, false, Bq,
                                                     (short)0, accQ, false, false);
    }

    // All D rows identical; row 0 (VGPR0, lanes 0..15) carries N = lane.
    if (lane < 16) {
        S[(size_t)b * strideT + t0 + n] = accS[0];
        Q[(size_t)b * strideT + t0 + n] = accQ[0];
    }
}

// ---------------------------------------------------------------------------
// Kernel 2: inclusive scan over T per batch (one block per b), converting
// (chan-sum, chan-sumsq) into (mean, rstd) IN PLACE.
// ---------------------------------------------------------------------------
__global__ void k_scan(float* __restrict__ S,   // in: colsum, out: mean
                       float* __restrict__ Q,   // in: colsumsq, out: rstd
                       int C, int T) {
    __shared__ float ss[256];
    __shared__ float sq[256];
    const int b   = blockIdx.x;
    const int tid = threadIdx.x;
    float* Sb = S + (size_t)b * T;
    float* Qb = Q + (size_t)b * T;

    float carryS = 0.0f, carryQ = 0.0f;
    const float fC = (float)C;

    for (int base = 0; base < T; base += 256) {
        const int t = base + tid;
        ss[tid] = (t < T) ? Sb[t] : 0.0f;
        sq[tid] = (t < T) ? Qb[t] : 0.0f;
        __syncthreads();

        // Hillis-Steele inclusive scan over the 256-chunk
        for (int off = 1; off < 256; off <<= 1) {
            float aS = 0.0f, aQ = 0.0f;
            if (tid >= off) { aS = ss[tid - off]; aQ = sq[tid - off]; }
            __syncthreads();
            ss[tid] += aS; sq[tid] += aQ;
            __syncthreads();
        }

        if (t < T) {
            const float cs   = carryS + ss[tid];
            const float cq   = carryQ + sq[tid];
            const float cnt  = (float)(t + 1) * fC;
            const float mean = cs / cnt;
            const float var  = cq / cnt - mean * mean;
            const float rstd = rsqrtf(var + EPSV);
            Sb[t] = mean;
            Qb[t] = rstd;
        }
        carryS += ss[255];
        carryQ += sq[255];
        __syncthreads();   // protect ss/sq before next chunk overwrites
    }
}

// ---------------------------------------------------------------------------
// Kernel 3: y = (x - mean[b,t]) * rstd[b,t] * w[c] + bias[c], float4 streams.
// grid = (ceil(T/4/256), C, B); c,b uniform per block -> scalar w/bias loads.
// ---------------------------------------------------------------------------
__global__ void k_norm(const float* __restrict__ x,
                       const float* __restrict__ weight,
                       const float* __restrict__ bias,
                       const float* __restrict__ mean,
                       const float* __restrict__ rstd,
                       float* __restrict__ y,
                       int C, int T) {
    const int c = blockIdx.y;
    const int b = blockIdx.z;
    const int t = (blockIdx.x * blockDim.x + threadIdx.x) << 2;
    if (t >= T) return;   // T is a multiple of 4 -> full float4 when t < T

    const float w  = weight[c];
    const float bb = bias[c];

    const size_t rowOff = ((size_t)b * C + c) * T + t;
    const size_t sOff   = (size_t)b * T + t;

    const float4 xv = *(const float4*)(x + rowOff);
    const float4 m4 = *(const float4*)(mean + sOff);
    const float4 r4 = *(const float4*)(rstd + sOff);

    float4 o;
    o.x = (xv.x - m4.x) * r4.x * w + bb;
    o.y = (xv.y - m4.y) * r4.y * w + bb;
    o.z = (xv.z - m4.z) * r4.z * w + bb;
    o.w = (xv.w - m4.w) * r4.w * w + bb;

    *(float4*)(y + rowOff) = o;
}

// ---------------------------------------------------------------------------
extern "C" void kernel_launch(void* const* d_in, const int* in_sizes, int n_in,
                              void* d_out, int out_size, void* d_ws, size_t ws_size,
                              hipStream_t stream) {
    const float* x      = (const float*)d_in[0];
    const float* weight = (const float*)d_in[1];
    const float* bias   = (const float*)d_in[2];
    float* y = (float*)d_out;

    const int B = 8;
    const int C = in_sizes[1];              // 256 (weight is 1 x C x 1)
    const int T = in_sizes[0] / (B * C);    // 16000

    // Workspace: S and Q, each B*T floats (1 MB total). Kernel 2 rewrites
    // them as mean and rstd in place.
    float* S = (float*)d_ws;
    float* Q = S + (size_t)B * T;

    // Kernel 1: B * (T/16) wave-tiles, 4 waves (128 threads) per block.
    const int tiles  = B * (T / 16);        // 8000
    const int blocks = tiles / 4;           // 2000 (exact)
    k_colsum_wmma<<<dim3(blocks), dim3(128), 0, stream>>>(x, S, Q, C, T);

    // Kernel 2: one block per batch element.
    k_scan<<<dim3(B), dim3(256), 0, stream>>>(S, Q, C, T);

    // Kernel 3: streaming normalize.
    const int tBlocks = (T / 4 + 255) / 256;  // 16
    k_norm<<<dim3(tBlocks, C, B), dim3(256), 0, stream>>>(x, weight, bias, S, Q, y, C, T);
}